// TaylorActivation_77043123355716
// MI455X (gfx1250) — compile-verified
//
#include <hip/hip_runtime.h>

// TaylorActivation: out = clip(poly8(clip(x, ±1e6)), ±1e6), elementwise.
// Memory-bound streaming kernel: 128-bit vector NT loads/stores, grid-stride.

#define CLIP_IN  1000000.0f
#define CLIP_OUT 1000000.0f

typedef float v4f __attribute__((ext_vector_type(4)));

__device__ __forceinline__ float poly_eval(float xv,
                                           float c0, float c1, float c2,
                                           float c3, float c4, float c5,
                                           float c6, float c7, float c8) {
    xv = fminf(fmaxf(xv, -CLIP_IN), CLIP_IN);
    float r = c8;
    r = fmaf(r, xv, c7);
    r = fmaf(r, xv, c6);
    r = fmaf(r, xv, c5);
    r = fmaf(r, xv, c4);
    r = fmaf(r, xv, c3);
    r = fmaf(r, xv, c2);
    r = fmaf(r, xv, c1);
    r = fmaf(r, xv, c0);
    return fminf(fmaxf(r, -CLIP_OUT), CLIP_OUT);
}

__global__ __launch_bounds__(256, 8)
void taylor_poly_v4_kernel(const float* __restrict__ x,
                           const float* __restrict__ w,
                           float* __restrict__ out,
                           long long n4,   // number of float4 chunks
                           long long n)    // total elements
{
    // Uniform-address readonly loads -> scalar (SMEM) loads into SGPRs.
    const float c0 = w[0], c1 = w[1], c2 = w[2], c3 = w[3], c4 = w[4];
    const float c5 = w[5], c6 = w[6], c7 = w[7], c8 = w[8];

    const v4f* __restrict__ x4 = reinterpret_cast<const v4f*>(x);
    v4f*       __restrict__ o4 = reinterpret_cast<v4f*>(out);

    const long long stride = (long long)gridDim.x * (long long)blockDim.x;
    long long i = (long long)blockIdx.x * blockDim.x + threadIdx.x;

    // Main vectorized stream: 16 B per lane per iteration, non-temporal
    // (single-touch data; don't pollute 192 MB L2 / WGP$).
    for (; i < n4; i += stride) {
        v4f v = __builtin_nontemporal_load(&x4[i]);
        v4f r;
        r.x = poly_eval(v.x, c0, c1, c2, c3, c4, c5, c6, c7, c8);
        r.y = poly_eval(v.y, c0, c1, c2, c3, c4, c5, c6, c7, c8);
        r.z = poly_eval(v.z, c0, c1, c2, c3, c4, c5, c6, c7, c8);
        r.w = poly_eval(v.w, c0, c1, c2, c3, c4, c5, c6, c7, c8);
        __builtin_nontemporal_store(r, &o4[i]);
    }

    // Scalar tail (n is 2^27 here so this is normally empty, but keep it
    // correct for any size).
    long long t = n4 * 4 + (long long)blockIdx.x * blockDim.x + threadIdx.x;
    for (; t < n; t += stride) {
        float v = __builtin_nontemporal_load(&x[t]);
        float r = poly_eval(v, c0, c1, c2, c3, c4, c5, c6, c7, c8);
        __builtin_nontemporal_store(r, &out[t]);
    }
}

extern "C" void kernel_launch(void* const* d_in, const int* in_sizes, int n_in,
                              void* d_out, int out_size, void* d_ws, size_t ws_size,
                              hipStream_t stream) {
    (void)n_in; (void)d_ws; (void)ws_size; (void)in_sizes;

    const float* x = (const float*)d_in[0];   // (16, 4096, 2048) fp32
    const float* w = (const float*)d_in[1];   // (9, 1) fp32 coefficients
    float* out = (float*)d_out;

    const long long n  = (long long)out_size;
    const long long n4 = n >> 2;

    const int block = 256;                    // 8 wave32 per block
    // ~8 float4 iterations per thread: plenty of waves to saturate HBM,
    // long per-wave streams for the memory system.
    long long want = (n4 + (long long)block * 8 - 1) / ((long long)block * 8);
    if (want < 1) want = 1;
    if (want > 1048576) want = 1048576;
    const int grid = (int)want;

    taylor_poly_v4_kernel<<<grid, block, 0, stream>>>(x, w, out, n4, n);
}